// NormalizedHistogram_34127810134625
// MI455X (gfx1250) — compile-verified
//
#include <hip/hip_runtime.h>

#define NBINS        256
#define NCH          3
#define NIMG         32
#define PIX          (512 * 512)                 // pixels per image
#define FLOATS_IMG   (PIX * NCH)                 // 786432 floats per image
#define VEC4_IMG     (FLOATS_IMG / 4)            // 196608 float4 per image
#define CHUNKS       96                          // blocks per image
#define VEC4_BLK     (VEC4_IMG / CHUNKS)         // 2048 float4 per block
#define THREADS      256
#define VEC4_THR     (VEC4_BLK / THREADS)        // 8 float4 per thread
#define NWAVES       (THREADS / 32)              // wave32 on gfx1250
#define HBINS        (NCH * NBINS)               // 768 counters per histogram
#define OUT_ELEMS    (NIMG * NBINS * NCH)        // 24576

__device__ __forceinline__ int bin_of(float v) {
    // bin_width = max(1/256, 1e-7) = 1/256 exactly; x/(1/256) == x*256.0f exactly
    int b = (int)(v * 256.0f);                   // trunc == floor for v >= 0
    b = b < 0 ? 0 : b;
    return b > (NBINS - 1) ? (NBINS - 1) : b;
}

__global__ void nh_zero_kernel(float* __restrict__ out, int n) {
    int i = blockIdx.x * blockDim.x + threadIdx.x;
    if (i < n) out[i] = 0.0f;
}

__global__ __launch_bounds__(THREADS)
void nh_hist_kernel(const float* __restrict__ x, float* __restrict__ out) {
    // Per-wave private histograms: 8 waves * 768 bins * 4B = 24 KB LDS (of 320 KB/WGP)
    __shared__ unsigned int lh[NWAVES][HBINS];

    const int tid  = threadIdx.x;
    const int wave = tid >> 5;                   // wave32
    const int img  = blockIdx.y;
    const int chnk = blockIdx.x;

    // Zero LDS sub-histograms with b128 stores: 1536 uint4 / 256 threads = 6 each
    uint4* l4 = (uint4*)&lh[0][0];
    #pragma unroll
    for (int j = 0; j < (NWAVES * HBINS / 4) / THREADS; ++j)
        l4[tid + j * THREADS] = make_uint4(0u, 0u, 0u, 0u);
    __syncthreads();

    const float4* __restrict__ xv = (const float4*)x;
    const unsigned int base = (unsigned int)img * VEC4_IMG
                            + (unsigned int)chnk * VEC4_BLK;
    unsigned int* h = lh[wave];

    // flat float index of .x = 4g; channel = 4g mod 3 = g mod 3 (4 == 1 mod 3).
    // Each unrolled step advances g by 256, and 256 == 1 (mod 3), so the
    // channel phase rotates by +1 per step: one modulo total, no per-iter div.
    int cph = (int)((base + (unsigned int)tid) % 3u);

    #pragma unroll
    for (int k = 0; k < VEC4_THR; ++k) {
        const unsigned int g = base + (unsigned int)(k * THREADS) + tid;
        const float4 v = xv[g];                  // global_load_b128, coalesced
        const int c0 = cph;
        int c1 = c0 + 1; if (c1 == 3) c1 = 0;
        int c2 = c1 + 1; if (c2 == 3) c2 = 0;
        // ds_add_u32 into this wave's private histogram
        atomicAdd(&h[c0 * NBINS + bin_of(v.x)], 1u);
        atomicAdd(&h[c1 * NBINS + bin_of(v.y)], 1u);
        atomicAdd(&h[c2 * NBINS + bin_of(v.z)], 1u);
        atomicAdd(&h[c0 * NBINS + bin_of(v.w)], 1u);
        cph = c1;                                // rotate phase for next step
    }
    __syncthreads();

    // Cross-wave reduce, then one float atomic per nonzero bin into [B,NBINS,C]
    for (int i = tid; i < HBINS; i += THREADS) {
        unsigned int s = 0u;
        #pragma unroll
        for (int w = 0; w < NWAVES; ++w) s += lh[w][i];
        if (s) {
            const int c   = i >> 8;              // i / NBINS
            const int bin = i & (NBINS - 1);
            // counts are integers < 2^24 -> float adds exact & order-independent
            atomicAdd(&out[(img * NBINS + bin) * NCH + c], (float)s);
        }
    }
}

__global__ __launch_bounds__(THREADS)
void nh_norm_kernel(float* __restrict__ out) {
    __shared__ float red[THREADS];
    const int bc  = blockIdx.x;                  // 0 .. NIMG*NCH-1
    const int img = bc / NCH;
    const int c   = bc % NCH;
    const int t   = threadIdx.x;                 // bin index

    const float v = out[(img * NBINS + t) * NCH + c];
    red[t] = v;
    __syncthreads();
    #pragma unroll
    for (int s = THREADS / 2; s > 0; s >>= 1) {
        if (t < s) red[t] += red[t + s];
        __syncthreads();
    }
    const float sum = red[0];                    // == 262144.0f (exact, pow2)
    out[(img * NBINS + t) * NCH + c] = v / sum;
}

extern "C" void kernel_launch(void* const* d_in, const int* in_sizes, int n_in,
                              void* d_out, int out_size, void* d_ws, size_t ws_size,
                              hipStream_t stream) {
    (void)in_sizes; (void)n_in; (void)d_ws; (void)ws_size; (void)out_size;
    const float* x = (const float*)d_in[0];
    float* out = (float*)d_out;

    // 1) zero the accumulator (d_out is poisoned by the harness)
    nh_zero_kernel<<<(OUT_ELEMS + THREADS - 1) / THREADS, THREADS, 0, stream>>>(out, OUT_ELEMS);

    // 2) histogram accumulation: 96 chunks x 32 images, 256 threads/block
    dim3 grid(CHUNKS, NIMG);
    nh_hist_kernel<<<grid, THREADS, 0, stream>>>(x, out);

    // 3) in-place normalization: one block per (image, channel)
    nh_norm_kernel<<<NIMG * NCH, THREADS, 0, stream>>>(out);
}